// FixedPointOperationForwardPass_44341242364266
// MI455X (gfx1250) — compile-verified
//
#include <hip/hip_runtime.h>
#include <hip/hip_bf16.h>
#include <math.h>

// Common-tangent (binodal) solver: analytic dG/d2G, fused dual-Newton inner
// loop, hardware v_log_f32 / v_rcp_f32, per-lane early exit.
// Transcendental/VALU-bound; no matmul structure -> WMMA not applicable.

#define EPS_X   1e-7f
#define RGAS    8.314f
#define N_OUTER 25
#define N_INNER 30
#define TOL_CT  1e-4f
#define TOL_NWT 1e-6f
#define LN2F    0.6931471805599453f

struct RKParams {
    float w0, w1, w2, w3;   // Omega0..3
    float g0;               // G0
    float rt;               // R*T
    float rtln2;            // R*T*ln(2)  (fold log2->ln conversion)
};

__device__ __forceinline__ float fast_rcp(float v) {
    return __builtin_amdgcn_rcpf(v);          // v_rcp_f32
}
__device__ __forceinline__ float log2_hw(float v) {
    return __builtin_amdgcn_logf(v);          // v_log_f32 (log base 2)
}

// G(x) = G0*x + RT*(x ln x + (1-x) ln(1-x)) + p*(w0 + w1 u + w2 u^2 + w3 u^3)
__device__ __forceinline__ float gibbs(float x, const RKParams& q) {
    float omx = 1.0f - x;
    float ent = q.rtln2 * (x * log2_hw(x) + omx * log2_hw(omx));
    float u   = 1.0f - 2.0f * x;
    float p   = x * omx;
    float rk  = p * fmaf(u, fmaf(u, fmaf(u, q.w3, q.w2), q.w1), q.w0);
    return fmaf(q.g0, x, ent + rk);
}

// dG/dx and d2G/dx2 at x (shared subexpressions).
__device__ __forceinline__ void dg_d2g(float x, const RKParams& q,
                                       float& dg, float& d2g) {
    float omx = 1.0f - x;
    float l   = log2_hw(x) - log2_hw(omx);          // log2(x/(1-x))
    float u   = 1.0f - 2.0f * x;
    float u2  = u * u;
    float u3  = u2 * u;
    float u4  = u2 * u2;
    float p   = x * omx;
    dg  = q.g0 + q.rtln2 * l
        + q.w0 * u
        + q.w1 * (u2 - 2.0f * p)
        + q.w2 * (u3 - 4.0f * p * u)
        + q.w3 * (u4 - 6.0f * p * u2);
    d2g = q.rt * fast_rcp(p)
        - 2.0f * q.w0
        - 6.0f * q.w1 * u
        + q.w2 * (2.0f - 12.0f * u2)
        + q.w3 * (6.0f * u - 20.0f * u3);
}

__device__ __forceinline__ float clamp01(float x) {
    return fminf(fmaxf(x, EPS_X), 1.0f - EPS_X);
}

__global__ void __launch_bounds__(256)
common_tangent_kernel(const float* __restrict__ x_in,
                      const float* __restrict__ params,
                      const float* __restrict__ Tp,
                      float* __restrict__ out, int B) {
    int i = blockIdx.x * blockDim.x + threadIdx.x;
    if (i >= B) return;

    RKParams q;                 // uniform loads -> s_load
    q.w0 = params[0];
    q.w1 = params[1];
    q.w2 = params[2];
    q.w3 = params[3];
    q.g0 = params[4];
    float T = Tp[0];
    q.rt    = RGAS * T;
    q.rtln2 = q.rt * LN2F;

    float xa = x_in[2 * i + 0];
    float xb = x_in[2 * i + 1];

    float ct  = (gibbs(xa, q) - gibbs(xb, q)) / (xa - xb);
    float dct = 9.99e6f;

    #pragma unroll 1
    for (int o = 0; o < N_OUTER; ++o) {
        if (dct <= TOL_CT) break;           // masked-update == no-op in ref

        // Two independent Newton chains fused for ILP (log/rcp co-exec).
        float xan = xa, xbn = xb;
        float dxa = 9.99e5f, dxb = 9.99e5f;
        #pragma unroll 1
        for (int k = 0; k < N_INNER; ++k) {
            bool aa = fabsf(dxa) > TOL_NWT;
            bool ab = fabsf(dxb) > TOL_NWT;
            if (!aa && !ab) break;
            float dga, d2a, dgb, d2b;
            dg_d2g(xan, q, dga, d2a);
            dg_d2g(xbn, q, dgb, d2b);
            if (aa) {
                dxa = (ct - dga) * fast_rcp(d2a);
                xan = clamp01(xan + dxa);
            }
            if (ab) {
                dxb = (ct - dgb) * fast_rcp(d2b);
                xbn = clamp01(xbn + dxb);
            }
        }

        float ctn = (gibbs(xan, q) - gibbs(xbn, q)) / (xan - xbn);
        dct = fabsf(ctn - ct);
        xa = xan;
        xb = xbn;
        ct = ctn;
    }

    out[2 * i + 0] = xa;
    out[2 * i + 1] = xb;
}

extern "C" void kernel_launch(void* const* d_in, const int* in_sizes, int n_in,
                              void* d_out, int out_size, void* d_ws, size_t ws_size,
                              hipStream_t stream) {
    (void)n_in; (void)d_ws; (void)ws_size; (void)out_size;
    const float* x      = (const float*)d_in[0];   // [B, 2]
    const float* params = (const float*)d_in[1];   // [5]
    const float* T      = (const float*)d_in[2];   // [1]
    float* out          = (float*)d_out;           // [B, 2]
    int B = in_sizes[0] / 2;
    int block = 256;
    int grid  = (B + block - 1) / block;
    common_tangent_kernel<<<grid, block, 0, stream>>>(x, params, T, out, B);
}